// GNN_39685497815503
// MI455X (gfx1250) — compile-verified
//
#include <hip/hip_runtime.h>
#include <hip/hip_bf16.h>
#include <stdint.h>

// ---------------------------------------------------------------------------
// GraphSAGE 2-layer forward for MI455X (gfx1250, wave32).
//   layer: h = relu( (segment_mean(x[src] -> dst)) @ W_l^T + b + x @ W_r^T )
// GEMM uses V_WMMA_F32_16X16X4_F32 (full f32, matches reference numerics).
// ---------------------------------------------------------------------------

#define N_NODES 100000
#define N_EDGES 1600000
#define DFEAT   128

typedef __attribute__((ext_vector_type(2))) float v2f;
typedef __attribute__((ext_vector_type(8))) float v8f;

// ---------------------------------------------------------------- zero fill
__global__ __launch_bounds__(256) void k_zero(float* __restrict__ p, long n) {
    long i = (long)blockIdx.x * 256 + threadIdx.x;
    if (i < n) p[i] = 0.0f;
}

// ------------------------------------------------------------------ degrees
__global__ __launch_bounds__(256) void k_degree(const long long* __restrict__ ei,
                                                float* __restrict__ deg) {
    int e = blockIdx.x * 256 + threadIdx.x;   // grid sized exactly: E % 256 == 0
    int d = (int)ei[(long)N_EDGES + e];
    atomicAdd(&deg[d], 1.0f);
}

// ------------------------------------------------- scatter-add of features
// 32 threads per edge; each handles one float4 chunk (coalesced gather +
// coalesced atomic adds). Source rows are L2-resident (51 MB < 192 MB L2).
__global__ __launch_bounds__(256) void k_scatter(const float* __restrict__ x,
                                                 const long long* __restrict__ ei,
                                                 float* __restrict__ agg) {
    long gid = (long)blockIdx.x * 256 + threadIdx.x;
    int e = (int)(gid >> 5);
    int c = (int)(gid & 31);
    int s = (int)ei[e];
    int d = (int)ei[(long)N_EDGES + e];
    const float4 v = reinterpret_cast<const float4*>(x)[(long)s * 32 + c];
    float* ap = agg + (long)d * DFEAT + c * 4;
    atomicAdd(ap + 0, v.x);
    atomicAdd(ap + 1, v.y);
    atomicAdd(ap + 2, v.z);
    atomicAdd(ap + 3, v.w);
}

// ------------------------------------- build Wt[k][o] = [W_l^T ; W_r^T]
// k in [0,128): W_l[o][k]; k in [128,256): W_r[o][k-128]. 256x128 floats.
__global__ __launch_bounds__(256) void k_wtrans(const float* __restrict__ Wl,
                                                const float* __restrict__ Wr,
                                                float* __restrict__ Wt) {
    int i = blockIdx.x * 256 + threadIdx.x;   // 32768 total, exact grid
    int k = i >> 7;
    int o = i & 127;
    Wt[i] = (k < 128) ? Wl[o * 128 + k] : Wr[o * 128 + (k - 128)];
}

// --------------------------------------------------- fused WMMA SAGE layer
// out[16 nodes x 128 feats] per block. 8 waves; wave w owns cols [16w,16w+16).
// A = [ agg/max(deg,1) | x ]  (16 x 256) staged in LDS, padded stride 260
// (260 % 64 == 4 -> the 16 lane-rows land on distinct LDS banks).
// B = Wt (256 x 128), lane-contiguous loads, hot in cache (reused by all
// 6250 blocks). Accumulate K=256 with 64 x v_wmma_f32_16x16x4_f32.
#define LDP 260
__global__ __launch_bounds__(256) void k_sage_gemm(const float* __restrict__ agg,
                                                   const float* __restrict__ xin,
                                                   const float* __restrict__ deg,
                                                   const float* __restrict__ Wt,
                                                   const float* __restrict__ bias,
                                                   float* __restrict__ out) {
    __shared__ float As[16 * LDP];
    __shared__ float invd[16];

    const int tid = threadIdx.x;
    const int nb  = blockIdx.x * 16;          // node base (100000/16 == 6250 exact)

    if (tid < 16) invd[tid] = 1.0f / fmaxf(deg[nb + tid], 1.0f);
    __syncthreads();

    // Stage A tile: cols [0,128) = normalized agg, cols [128,256) = x.
    for (int i = tid; i < 16 * 256; i += 256) {
        int row = i >> 8;
        int k   = i & 255;
        float v = (k < 128)
                    ? agg[(long)(nb + row) * DFEAT + k] * invd[row]
                    : xin[(long)(nb + row) * DFEAT + (k - 128)];
        As[row * LDP + k] = v;
    }
    __syncthreads();

    const int lane  = tid & 31;
    const int wave  = tid >> 5;
    const int col   = wave * 16;
    const int n     = lane & 15;      // output column within tile / A row (M)
    const int hi    = lane >> 4;      // lane half selects K pair and M half
    const int khalf = hi << 1;        // 0 or 2

    const float* arow = &As[n * LDP];

    v8f acc = {};
#pragma unroll 4
    for (int kk = 0; kk < 256; kk += 4) {
        const int k0 = kk + khalf;
        v2f a; a.x = arow[k0];
               a.y = arow[k0 + 1];
        v2f b; b.x = Wt[k0 * 128 + col + n];
               b.y = Wt[(k0 + 1) * 128 + col + n];
        acc = __builtin_amdgcn_wmma_f32_16x16x4_f32(
                  false, a, false, b, (short)0, acc, false, false);
    }

    // Epilogue: bias + ReLU, C/D layout: VGPR r -> M = r + 8*hi, N = lane&15.
    const float bv    = bias[col + n];
    const int   mbase = hi << 3;
#pragma unroll
    for (int r = 0; r < 8; ++r) {
        float v = acc[r] + bv;
        v = fmaxf(v, 0.0f);
        out[(long)(nb + mbase + r) * DFEAT + col + n] = v;
    }
}

// ---------------------------------------------------------------------------
extern "C" void kernel_launch(void* const* d_in, const int* in_sizes, int n_in,
                              void* d_out, int out_size, void* d_ws, size_t ws_size,
                              hipStream_t stream) {
    const float*     x   = (const float*)d_in[0];
    const long long* ei  = (const long long*)d_in[1];   // int64 edge_index [2,E]
    const float*     W1l = (const float*)d_in[2];
    const float*     W1r = (const float*)d_in[3];
    const float*     b1  = (const float*)d_in[4];
    const float*     W2l = (const float*)d_in[5];
    const float*     W2r = (const float*)d_in[6];
    const float*     b2  = (const float*)d_in[7];
    float*           out = (float*)d_out;

    // Workspace carve-up (bytes): deg | agg | h1 | Wt
    char* ws = (char*)d_ws;
    float* deg = (float*)(ws);                                  // N floats
    float* agg = (float*)(ws + (1u << 20));                     // N*128 floats
    float* h1  = (float*)(ws + (1u << 20) + (size_t)N_NODES * DFEAT * 4);
    float* Wt  = (float*)(ws + (1u << 20) + 2 * (size_t)N_NODES * DFEAT * 4);

    const long nfeat = (long)N_NODES * DFEAT;                   // 12.8M
    const int  zb_feat = (int)((nfeat + 255) / 256);
    const int  zb_deg  = (N_NODES + 255) / 256;
    const int  eb      = N_EDGES / 256;                         // 6250 (exact)
    const int  sb      = (int)(((long)N_EDGES * 32) / 256);     // 200000 (exact)
    const int  gb      = N_NODES / 16;                          // 6250 (exact)

    // ---- shared degree (same graph both layers) ----
    k_zero<<<zb_deg, 256, 0, stream>>>(deg, N_NODES);
    k_degree<<<eb, 256, 0, stream>>>(ei, deg);

    // ---- layer 1 ----
    k_zero<<<zb_feat, 256, 0, stream>>>(agg, nfeat);
    k_scatter<<<sb, 256, 0, stream>>>(x, ei, agg);
    k_wtrans<<<128, 256, 0, stream>>>(W1l, W1r, Wt);
    k_sage_gemm<<<gb, 256, 0, stream>>>(agg, x, deg, Wt, b1, h1);

    // ---- layer 2 ----
    k_zero<<<zb_feat, 256, 0, stream>>>(agg, nfeat);
    k_scatter<<<sb, 256, 0, stream>>>(h1, ei, agg);
    k_wtrans<<<128, 256, 0, stream>>>(W2l, W2r, Wt);
    k_sage_gemm<<<gb, 256, 0, stream>>>(agg, h1, deg, Wt, b2, out);
}